// PointConvolution_32306744000737
// MI455X (gfx1250) — compile-verified
//
#include <hip/hip_runtime.h>

typedef __attribute__((ext_vector_type(2))) float v2f;
typedef __attribute__((ext_vector_type(8))) float v8f;

static constexpr int BATCH = 16;
static constexpr int CIN   = 3;
static constexpr int OUTC  = 32;
static constexpr int H     = 512;
static constexpr int Wd    = 512;
static constexpr int KF    = 24;          // 8 offsets * 3 channels
static constexpr int HW    = H * Wd;
static constexpr int TILES_PER_WAVE = 16; // half a row: 256 px

// COORDS order from the reference: (i,j) pairs minus center, (di,dj)=(i-1,j-1)
static constexpr int DI[8] = {-1, 0, 1, -1, 1, -1, 0, 1};
static constexpr int DJ[8] = {-1,-1,-1,  0, 0,  1, 1, 1};

// Per-wave (h-dependent) precompute for one B element.
// KLO used by lanes 0-15, KHI by lanes 16-31 (B-matrix K = v + 2*(lane>=16)).
template<int KLO, int KHI>
__device__ __forceinline__ void prep(int xb, int h, bool hiHalf,
                                     unsigned& rowoff, int& dj, bool& vrow) {
  constexpr int dil = DI[KLO / 3], djl = DJ[KLO / 3], chl = KLO % 3;
  constexpr int dih = DI[KHI / 3], djh = DJ[KHI / 3], chh = KHI % 3;
  const int di = hiHalf ? dih : dil;
  const int ch = hiHalf ? chh : chl;
  dj = hiHalf ? djh : djl;
  const int hh = h + di;
  vrow = (unsigned)hh < (unsigned)H;
  const int hc = hh < 0 ? 0 : (hh > H - 1 ? H - 1 : hh);
  rowoff = (unsigned)((xb + ch) * HW + hc * Wd);   // element offset of row start
}

// Branchless feat value: clamped load + select, EXEC stays all-ones.
template<int KLO, int KHI>
__device__ __forceinline__ float featv(const float* __restrict__ x,
                                       unsigned rowoff, int dj, bool vrow,
                                       int w, bool hiHalf, const float cx[3]) {
  constexpr int chl = KLO % 3, chh = KHI % 3;
  const float c  = hiHalf ? cx[chh] : cx[chl];
  const int   ww = w + dj;
  const bool  ok = vrow && ((unsigned)ww < (unsigned)Wd);
  const int   wc = ww < 0 ? 0 : (ww > Wd - 1 ? Wd - 1 : ww);
  float v = x[rowoff + (unsigned)wc];
  v = ok ? v : 0.0f;
  return v - c;
}

__global__ __launch_bounds__(256)
void PointConvolution_32306744000737_kernel(const float* __restrict__ x,
                                            const float* __restrict__ Wm,
                                            const float* __restrict__ bias,
                                            float* __restrict__ out) {
  const int  lane = threadIdx.x & 31;
  const int  wave = (int)((blockIdx.x * blockDim.x + threadIdx.x) >> 5); // 0..16383
  const int  n    = lane & 15;        // pixel / M-row within tile
  const bool hi   = lane >= 16;       // upper lane half -> K+2 / M+8
  const int  hk   = hi ? 2 : 0;

  // Each wave owns half an image row: fixed (b, h), 16 tiles of 16 px.
  const int b  = wave >> 10;                  // 1024 half-rows per image
  const int h  = (wave >> 1) & (H - 1);
  const int w0 = (wave & 1) << 8;             // 0 or 256
  const int xb = b * CIN;

  // ---- A matrix: W[32x24], resident in registers across all tiles ----
  // A layout (16x4 f32): lane%16 = M, elem v holds K = v + 2*(lane>=16)
  v2f a0[6], a1[6];
#pragma unroll
  for (int kk = 0; kk < 6; ++kk) {
    const int kb = 4 * kk + hk;
    a0[kk].x = Wm[n * KF + kb];
    a0[kk].y = Wm[n * KF + kb + 1];
    a1[kk].x = Wm[(16 + n) * KF + kb];
    a1[kk].y = Wm[(16 + n) * KF + kb + 1];
  }

  // ---- bias folded into accumulator init (D: M = v + 8*(lane>=16)) ----
  v8f cinit0, cinit1;
#pragma unroll
  for (int v = 0; v < 8; ++v) {
    cinit0[v] = bias[v + (hi ? 8 : 0)];
    cinit1[v] = bias[16 + v + (hi ? 8 : 0)];
  }

  // ---- hoist all h-dependent work for the 12 B elements out of the loop ----
  unsigned ro[6][2];
  int      dj[6][2];
  bool     vr[6][2];
#define PREP(KB)                                                               \
  prep<KB + 0, KB + 2>(xb, h, hi, ro[KB / 4][0], dj[KB / 4][0], vr[KB / 4][0]);\
  prep<KB + 1, KB + 3>(xb, h, hi, ro[KB / 4][1], dj[KB / 4][1], vr[KB / 4][1]);
  PREP(0) PREP(4) PREP(8) PREP(12) PREP(16) PREP(20)
#undef PREP

  const unsigned cx0  = (unsigned)(xb * HW + h * Wd);          // center row, ch 0
  const unsigned outE = (unsigned)(b * OUTC * HW + h * Wd);    // out row base
  const int      ob   = hi ? 8 : 0;

  // Second uniform store base (SGPR pair) so both 8-store clauses use
  // 24-bit immediate channel offsets (0..7*HW*4 bytes) off one lane offset.
  float* __restrict__ out1 = out + (size_t)16 * HW;

  int w = w0 + n;
#pragma unroll 1
  for (int it = 0; it < TILES_PER_WAVE; ++it, w += 16) {
    float cx[3];
    cx[0] = x[cx0 + (unsigned)w];
    cx[1] = x[cx0 + HW + (unsigned)w];
    cx[2] = x[cx0 + 2 * HW + (unsigned)w];

    v8f acc0 = cinit0;
    v8f acc1 = cinit1;

#define KSTEP(KB)                                                              \
    {                                                                          \
      v2f bv;                                                                  \
      bv.x = featv<KB + 0, KB + 2>(x, ro[KB / 4][0], dj[KB / 4][0],            \
                                   vr[KB / 4][0], w, hi, cx);                  \
      bv.y = featv<KB + 1, KB + 3>(x, ro[KB / 4][1], dj[KB / 4][1],            \
                                   vr[KB / 4][1], w, hi, cx);                  \
      acc0 = __builtin_amdgcn_wmma_f32_16x16x4_f32(false, a0[KB / 4], false,   \
                 bv, (short)0, acc0, false, false);                            \
      acc1 = __builtin_amdgcn_wmma_f32_16x16x4_f32(false, a1[KB / 4], false,   \
                 bv, (short)0, acc1, false, false);                            \
    }
    KSTEP(0) KSTEP(4) KSTEP(8) KSTEP(12) KSTEP(16) KSTEP(20)
#undef KSTEP

    // ---- store: D[v][lane] = out[b, obase+v, h, w] ----
    const unsigned o0 = outE + (unsigned)w + (unsigned)(ob * HW);
#pragma unroll
    for (int v = 0; v < 8; ++v) {
      out[o0 + (unsigned)(v * HW)]  = acc0[v];
      out1[o0 + (unsigned)(v * HW)] = acc1[v];
    }
  }
}

extern "C" void kernel_launch(void* const* d_in, const int* in_sizes, int n_in,
                              void* d_out, int out_size, void* d_ws, size_t ws_size,
                              hipStream_t stream) {
  const float* x    = (const float*)d_in[0];
  const float* Wm   = (const float*)d_in[1];
  const float* bias = (const float*)d_in[2];
  float*       out  = (float*)d_out;

  // 16384 waves (2048 blocks * 8 waves) x 16 tiles = 262144 tiles, exact cover.
  const int threads = 256;
  const int blocks  = (BATCH * H * 2) / (threads / 32);   // 2048

  PointConvolution_32306744000737_kernel<<<dim3(blocks), dim3(threads), 0, stream>>>(
      x, Wm, bias, out);
}